// RetNetRetention_48902497632661
// MI455X (gfx1250) — compile-verified
//
#include <hip/hip_runtime.h>

// RetNet-style elementwise retention scan:
//   r_t = 0.9 * r_{t-1} + k_t * v_t ;  out_t = q_t * r_t   (per (b,d) channel)
//
// Memory-bound (512 MB min traffic). Implemented as a chunked parallel scan:
//   Pass 1: per-chunk zero-init final states T_c (reads k,v).
//   Pass 2: exact carry-in combine + intra-chunk scan done 16 timesteps at a
//           time as R = A*X with V_WMMA_F32_16X16X4_F32, where A is the
//           constant lower-triangular decay matrix A[t,s] = 0.9^(t-s).

typedef __attribute__((ext_vector_type(2))) float v2f;
typedef __attribute__((ext_vector_type(8))) float v8f;

#define DECAY_F 0.9f

constexpr int Bc = 8;
constexpr int Sc = 4096;
constexpr int Dc = 1024;
constexpr int NC = 16;        // time chunks
constexpr int Lc = Sc / NC;   // 256 steps per chunk

// ---------------------------------------------------------------------------
// Pass 1: each thread owns one (b, d, chunk); computes the chunk's final state
// with zero initial condition:  T = sum_{i<L} 0.9^(L-1-i) * k_i * v_i.
// Threads are laid out with d contiguous -> fully coalesced 128B/wave streams.
// ---------------------------------------------------------------------------
__global__ void __launch_bounds__(256)
retnet_pass1_chunk_carries(const float* __restrict__ k,
                           const float* __restrict__ v,
                           float* __restrict__ ws) {
  const int gt   = blockIdx.x * blockDim.x + threadIdx.x;  // Bc*NC*Dc threads
  const int d    = gt % Dc;
  const int rest = gt / Dc;
  const int c    = rest % NC;
  const int b    = rest / NC;

  const size_t base = ((size_t)b * Sc + (size_t)c * Lc) * Dc + (size_t)d;
  const float* kp = k + base;
  const float* vp = v + base;

  float st = 0.0f;
#pragma unroll 4
  for (int t = 0; t < Lc; ++t) {
    st = fmaf(DECAY_F, st, kp[(size_t)t * Dc] * vp[(size_t)t * Dc]);
  }
  ws[((size_t)b * NC + c) * Dc + d] = st;
}

// ---------------------------------------------------------------------------
// Pass 2: one wave per (b, 16-channel tile, chunk). For each 16-timestep
// block: X[s,ch] = k*v, R = A*X via 4x v_wmma_f32_16x16x4_f32, then
// r = R + a^(t+1)*carry, out = q*r, and the carry is the t=15 row.
//
// Fragment layouts (wave32, f32):
//   A (16x4):  lane&15 = M row; VGPR0 = K{0|2}, VGPR1 = K{1|3} (lo|hi half)
//   B (4x16):  lanes = N (channel); VGPR0 = K{0|2}, VGPR1 = K{1|3}
//   C/D:       lanes = N; VGPR j = M row j (lo half) / j+8 (hi half)
// ---------------------------------------------------------------------------
__global__ void __launch_bounds__(256)
retnet_pass2_wmma(const float* __restrict__ q,
                  const float* __restrict__ k,
                  const float* __restrict__ v,
                  const float* __restrict__ ws,
                  float* __restrict__ out) {
  const int tid  = threadIdx.x;
  const int wave = (blockIdx.x * 256 + tid) >> 5;
  const int lane = tid & 31;
  const int ch   = lane & 15;   // channel within tile; also A row M
  const int hi   = lane >> 4;   // 0: lanes 0-15, 1: lanes 16-31

  const int c    = wave % NC;                    // time chunk (uniform in wave)
  const int tile = wave / NC;
  const int d0   = (tile & (Dc / 16 - 1)) << 4;  // tile base channel
  const int b    = tile / (Dc / 16);
  const int d    = d0 + ch;

  // Decay powers 0.9^0 .. 0.9^16
  float pw[17];
  pw[0] = 1.0f;
#pragma unroll
  for (int i = 1; i <= 16; ++i) pw[i] = pw[i - 1] * DECAY_F;

  // A-matrix fragments: A[m,s] = (m>=s) ? 0.9^(m-s) : 0, K split in 4 chunks.
  v2f af[4];
#pragma unroll
  for (int kk = 0; kk < 4; ++kk) {
    const int s0 = 4 * kk + 2 * hi;
    af[kk].x = (ch >= s0)     ? pw[ch - s0]     : 0.0f;
    af[kk].y = (ch >= s0 + 1) ? pw[ch - s0 - 1] : 0.0f;
  }

  // carry scale per D-register: r[t] gets a^(t+1) * carry, t = j + 8*hi
  float cs[8];
#pragma unroll
  for (int j = 0; j < 8; ++j) cs[j] = pw[j + 1 + 8 * hi];

  // a^Lc = (a^16)^(Lc/16)  (~2e-12 for Lc=256; still normal fp32)
  float pL = pw[16];
#pragma unroll
  for (int i = 1; i < Lc / 16; ++i) pL *= pw[16];

  // Exact carry-in for this chunk: R = a^L * R + T_{c'} over preceding chunks.
  float carry = 0.0f;
  {
    const float* wsp = ws + (size_t)b * NC * Dc + (size_t)d;
    for (int cp = 0; cp < c; ++cp)
      carry = fmaf(pL, carry, wsp[(size_t)cp * Dc]);
  }

  const size_t base = ((size_t)b * Sc + (size_t)c * Lc) * Dc + (size_t)d;
  const float* qb = q + base;
  const float* kb = k + base;
  const float* vb = v + base;
  float*       ob = out + base;

  for (int ib = 0; ib < Lc / 16; ++ib) {
    const size_t off = (size_t)ib * 16 * Dc;

    v8f acc = {};
#pragma unroll
    for (int kk = 0; kk < 4; ++kk) {
      const int s0 = 4 * kk + 2 * hi;       // B-fragment K rows for this lane
      v2f xb;
      xb.x = kb[off + (size_t)s0 * Dc]       * vb[off + (size_t)s0 * Dc];
      xb.y = kb[off + (size_t)(s0 + 1) * Dc] * vb[off + (size_t)(s0 + 1) * Dc];
      // D = A * X + C   (8 args: neg_a, A, neg_b, B, c_mod, C, reuse_a, reuse_b)
      acc = __builtin_amdgcn_wmma_f32_16x16x4_f32(
          false, af[kk], false, xb, (short)0, acc, false, false);
    }

    float rlast = 0.0f;
#pragma unroll
    for (int j = 0; j < 8; ++j) {
      const int tj  = j + 8 * hi;                    // timestep of this D reg
      const float r = fmaf(cs[j], carry, acc[j]);    // add decayed carry
      ob[off + (size_t)tj * Dc] = qb[off + (size_t)tj * Dc] * r;
      if (j == 7) rlast = r;                         // state at t0+7 / t0+15
    }
    // New carry = state at local t=15, held in hi half's register 7.
    carry = __shfl(rlast, 16 + ch, 32);
  }
}

// ---------------------------------------------------------------------------
// Launcher. Inputs (setup_inputs order): q, k, v — all fp32 (B,S,D).
// ---------------------------------------------------------------------------
extern "C" void kernel_launch(void* const* d_in, const int* in_sizes, int n_in,
                              void* d_out, int out_size, void* d_ws, size_t ws_size,
                              hipStream_t stream) {
  (void)in_sizes; (void)n_in; (void)out_size; (void)ws_size;
  const float* q = (const float*)d_in[0];
  const float* k = (const float*)d_in[1];
  const float* v = (const float*)d_in[2];
  float* out = (float*)d_out;
  float* ws  = (float*)d_ws;   // Bc*NC*Dc floats = 512 KB of chunk carries

  // Pass 1: Bc*NC*Dc threads (4096 waves), d contiguous per thread.
  const int threads1 = Bc * NC * Dc;
  retnet_pass1_chunk_carries<<<threads1 / 256, 256, 0, stream>>>(k, v, ws);

  // Pass 2: one wave per (b, 16-ch tile, chunk) = 8192 waves.
  const int waves2 = Bc * (Dc / 16) * NC;
  retnet_pass2_wmma<<<waves2 * 32 / 256, 256, 0, stream>>>(q, k, v, ws, out);
}